// IOTransformer_1760936591416
// MI455X (gfx1250) — compile-verified
//
#include <hip/hip_runtime.h>
#include <math.h>

typedef float v2f __attribute__((ext_vector_type(2)));
typedef float v8f __attribute__((ext_vector_type(8)));

__device__ __forceinline__ float softplusf(float x) {
    return (x > 20.f) ? x : log1pf(expf(x));
}

// ---------------------------------------------------------------------------
// Kernel 1: fused head GEMM.
//   out[row, n] = sum_d h[row,d] * (W[n,d] + sp(tied_scale)*E[5+n,d]) + bias[n]
// M = B*T = 16384, K = 1024, N = 96.
// Block: 256 threads (8 waves), BM=128, BK=32, double-buffered LDS.
// Wave layout: wave w owns rows [(w>>1)*32, +32) x cols [(w&1)*48, +48)
//   -> 6 v8f accumulators, 2 A-fragments shared across 3 B-fragments.
// A tile is staged with GLOBAL_LOAD_ASYNC_TO_LDS_B128 (ASYNCcnt) and
// double-buffered so async loads of chunk c+1 overlap WMMA on chunk c.
// B tile is stored pair-interleaved so each WMMA B-fragment is one ds_load_b64.
// ---------------------------------------------------------------------------
#define BM 128
#define BK 32
#define LDA 36                 // padded A stride (floats)
#define BPAIR (96 * 2)         // floats per k-pair row of B tile
#define NC (1024 / BK)         // 32 k-chunks

__global__ __launch_bounds__(256) void head_gemm_kernel(
    const float* __restrict__ h,  const float* __restrict__ E,
    const float* __restrict__ Wn, const float* __restrict__ bn,
    const float* __restrict__ Wt, const float* __restrict__ bt,
    const float* __restrict__ tsa, const float* __restrict__ tst,
    float* __restrict__ out_act, float* __restrict__ out_time)
{
    __shared__ float As[2][BM * LDA];           // 2 x 18432 B
    __shared__ float Bs[2][(BK / 2) * BPAIR];   // 2 x 12288 B

    const int tid    = threadIdx.x;
    const int wave   = tid >> 5;
    const int lane   = tid & 31;
    const int mlan   = lane & 15;          // m (A) or n (B/D) within a 16x16 tile
    const int kp     = (lane >> 4) << 1;   // K sub-offset: 0 or 2
    const int mrow0  = (wave >> 1) * 32;   // wave's local 32-row strip
    const int nhalf  = wave & 1;           // wave's 48-col half
    const int blockM = blockIdx.x * BM;

    const float spa = softplusf(tsa[0]);
    const float spt = softplusf(tst[0]);

    v8f acc[6];
    #pragma unroll
    for (int i = 0; i < 6; ++i)
        #pragma unroll
        for (int j = 0; j < 8; ++j) acc[i][j] = 0.f;

    // ---- A async loader: 2 threads per row, 16 floats (4 x b128) each ----
    const int rowA  = tid >> 1;
    const int halfA = tid & 1;
    const unsigned aGlbBase =
        (unsigned)(((blockM + rowA) * 1024 + halfA * 16) * 4); // bytes (<2^31)

    auto loadA = [&](int kc, int buf) {
        const unsigned lds0 =
            (unsigned)(uintptr_t)&As[buf][rowA * LDA + halfA * 16];
        const unsigned g0 = aGlbBase + (unsigned)(kc * 4);
        #pragma unroll
        for (int i = 0; i < 4; ++i) {
            asm volatile("global_load_async_to_lds_b128 %0, %1, %2"
                         :: "v"(lds0 + i * 16), "v"(g0 + i * 16), "s"(h)
                         : "memory");
        }
    };

    // ---- B loader: fused weight combine, pair-interleaved LDS layout ----
    auto loadB = [&](int kc, int buf) {
        #pragma unroll
        for (int jj = 0; jj < 3; ++jj) {
            const int j  = tid + jj * 256;       // 0..767
            const int n  = j >> 3;               // 0..95
            const int k4 = (j & 7) << 2;         // 0..28 step 4
            const float* erow = E + (size_t)(5 + n) * 1024 + kc + k4;
            const float* wrow = ((n < 64) ? (Wn + (size_t)n * 1024)
                                          : (Wt + (size_t)(n - 64) * 1024)) + kc + k4;
            const float scale = (n < 64) ? spa : spt;
            const float4 wv = *(const float4*)wrow;
            const float4 ev = *(const float4*)erow;
            float2 p0; p0.x = wv.x + scale * ev.x; p0.y = wv.y + scale * ev.y;
            float2 p1; p1.x = wv.z + scale * ev.z; p1.y = wv.w + scale * ev.w;
            float* dst = &Bs[buf][(k4 >> 1) * BPAIR + n * 2];
            *(float2*)dst           = p0;   // pair (k4, k4+1)
            *(float2*)(dst + BPAIR) = p1;   // pair (k4+2, k4+3)
        }
    };

    loadA(0, 0);
    loadB(0, 0);
    asm volatile("s_wait_asynccnt 0x0" ::: "memory");
    __syncthreads();

    #pragma unroll 1
    for (int c = 0; c < NC; ++c) {
        const int cur = c & 1;
        if (c + 1 < NC) {           // prefetch next chunk into the other buffer
            loadA((c + 1) * BK, cur ^ 1);
            loadB((c + 1) * BK, cur ^ 1);
        }
        const float* aB = &As[cur][(mrow0 + mlan) * LDA];
        const float* bB = &Bs[cur][(nhalf * 48 + mlan) * 2];
        #pragma unroll
        for (int ks = 0; ks < BK; ks += 4) {
            const v2f a0 = *(const v2f*)(aB + ks + kp);              // rows m..+15
            const v2f a1 = *(const v2f*)(aB + 16 * LDA + ks + kp);   // rows +16..+31
            const int pr = ((ks + kp) >> 1) * BPAIR;
            #pragma unroll
            for (int nt = 0; nt < 3; ++nt) {
                const v2f b = *(const v2f*)(bB + pr + nt * 32);      // one b64
                acc[nt] = __builtin_amdgcn_wmma_f32_16x16x4_f32(
                    false, a0, false, b, (short)0, acc[nt], false, false);
                acc[3 + nt] = __builtin_amdgcn_wmma_f32_16x16x4_f32(
                    false, a1, false, b, (short)0, acc[3 + nt], false, false);
            }
        }
        asm volatile("s_wait_asynccnt 0x0" ::: "memory");
        __syncthreads();
    }

    // ---- epilogue: D layout m = r + 8*(lane>=16), n = lane%16 ----
    const int rowBase = blockM + mrow0 + (lane >> 4) * 8;
    #pragma unroll
    for (int mt = 0; mt < 2; ++mt) {
        #pragma unroll
        for (int nt = 0; nt < 3; ++nt) {
            const int col = nhalf * 48 + nt * 16 + mlan;
            #pragma unroll
            for (int r = 0; r < 8; ++r) {
                const int   row = rowBase + mt * 16 + r;
                const float v   = acc[mt * 3 + nt][r];
                if (col < 64) out_act [(size_t)row * 64 + col]        = v + bn[col];
                else          out_time[(size_t)row * 32 + (col - 64)] = v + bt[col - 64];
            }
        }
    }
}

// ---------------------------------------------------------------------------
// Kernel 2: sparse retrieval/copy head (unchanged; sparse -> latency-trivial).
// One block per (b, q). Exits unless tokens[b,q]==<LABEL>(2). Builds compacted
// valid-key lists in LDS, one wave per key computes the normalized 1024-dim
// dot, buckets tau*sim per class, then adds sp(copy_scale)*bucket into the
// logits written by kernel 1 (same stream => ordered, deterministic).
// ---------------------------------------------------------------------------
__global__ __launch_bounds__(256) void copy_head_kernel(
    const int* __restrict__ tokens, const float* __restrict__ h,
    const float* __restrict__ csa, const float* __restrict__ cst,
    const float* __restrict__ cta, const float* __restrict__ ctt,
    float* __restrict__ out_act, float* __restrict__ out_time)
{
    const int q    = blockIdx.x;
    const int b    = blockIdx.y;
    const int T    = gridDim.x;
    const int base = b * T;
    if (tokens[base + q] != 2) return;   // uniform branch: whole block exits

    __shared__ float hq[1024];
    __shared__ float buckets[96];
    __shared__ float qn2s;
    __shared__ int   nact, ntim;
    __shared__ int   ka[2048];
    __shared__ int   kt[2048];

    const int tid = threadIdx.x;
    if (tid == 0) { nact = 0; ntim = 0; qn2s = 0.f; }
    if (tid < 96) buckets[tid] = 0.f;
    __syncthreads();

    { // stage query row + its squared norm
        const float4 v = *(const float4*)(h + (size_t)(base + q) * 1024 + tid * 4);
        *(float4*)&hq[tid * 4] = v;
        atomicAdd(&qn2s, v.x * v.x + v.y * v.y + v.z * v.z + v.w * v.w);
    }
    // compact valid keys: strict past, prev token == <LABEL>, class in range
    for (int k = tid; k < q; k += 256) {
        if (k >= 1 && tokens[base + k - 1] == 2) {
            const int tok = tokens[base + k];
            if (tok >= 5 && tok < 69) {
                const int idx = atomicAdd(&nact, 1); ka[idx] = (k << 8) | (tok - 5);
            } else if (tok >= 69 && tok < 101) {
                const int idx = atomicAdd(&ntim, 1); kt[idx] = (k << 8) | (tok - 69);
            }
        }
    }
    __syncthreads();

    const float invq  = 1.f / fmaxf(sqrtf(qn2s), 1e-12f);
    const float tau_a = softplusf(cta[0]);
    const float tau_t = softplusf(ctt[0]);
    const int wave = tid >> 5, lane = tid & 31;
    const int na = nact, ntc = ntim;

    for (int i = wave; i < na + ntc; i += 8) {
        const bool isAct  = (i < na);
        const int  packed = isAct ? ka[i] : kt[i - na];
        const int  k = packed >> 8, cls = packed & 255;
        const float* hk = h + (size_t)(base + k) * 1024;
        float dot = 0.f, nk2 = 0.f;
        #pragma unroll
        for (int it = 0; it < 8; ++it) {
            const int d = it * 128 + lane * 4;          // coalesced 512B per iter
            const float4 kv = *(const float4*)(hk + d);
            const float4 qv = *(const float4*)&hq[d];
            dot += kv.x * qv.x + kv.y * qv.y + kv.z * qv.z + kv.w * qv.w;
            nk2 += kv.x * kv.x + kv.y * kv.y + kv.z * kv.z + kv.w * kv.w;
        }
        #pragma unroll
        for (int off = 16; off > 0; off >>= 1) {
            dot += __shfl_xor(dot, off, 32);
            nk2 += __shfl_xor(nk2, off, 32);
        }
        if (lane == 0) {
            const float sim = dot * invq / fmaxf(sqrtf(nk2), 1e-12f);
            if (isAct) atomicAdd(&buckets[cls],      tau_a * sim);
            else       atomicAdd(&buckets[64 + cls], tau_t * sim);
        }
    }
    __syncthreads();

    const float sca = softplusf(csa[0]);
    const float sct = softplusf(cst[0]);
    if (tid < 64)      out_act [(size_t)(base + q) * 64 + tid]        += sca * buckets[tid];
    else if (tid < 96) out_time[(size_t)(base + q) * 32 + (tid - 64)] += sct * buckets[tid];
}

// ---------------------------------------------------------------------------
extern "C" void kernel_launch(void* const* d_in, const int* in_sizes, int n_in,
                              void* d_out, int out_size, void* d_ws, size_t ws_size,
                              hipStream_t stream) {
    (void)in_sizes; (void)n_in; (void)out_size; (void)d_ws; (void)ws_size;
    const int* tokens = (const int*)  d_in[0];
    const float* h    = (const float*)d_in[1];
    const float* E    = (const float*)d_in[2];
    const float* Wn   = (const float*)d_in[3];
    const float* bn   = (const float*)d_in[4];
    const float* Wt   = (const float*)d_in[5];
    const float* bt   = (const float*)d_in[6];
    const float* tsa  = (const float*)d_in[7];
    const float* tst  = (const float*)d_in[8];
    const float* csa  = (const float*)d_in[9];
    const float* cst  = (const float*)d_in[10];
    const float* cta  = (const float*)d_in[11];
    const float* ctt  = (const float*)d_in[12];

    const int B = 8, T = 2048;
    float* out_act  = (float*)d_out;
    float* out_time = out_act + (size_t)B * T * 64;

    head_gemm_kernel<<<dim3((B * T) / BM), 256, 0, stream>>>(
        h, E, Wn, bn, Wt, bt, tsa, tst, out_act, out_time);
    copy_head_kernel<<<dim3(T, B), 256, 0, stream>>>(
        tokens, h, csa, cst, cta, ctt, out_act, out_time);
}